// RerankingDecoder_14405320311457
// MI455X (gfx1250) — compile-verified
//
#include <hip/hip_runtime.h>
#include <math.h>
#include <stdint.h>

typedef __attribute__((ext_vector_type(2))) float v2f;
typedef __attribute__((ext_vector_type(8))) float v8f;

#define HIDDEN  256
#define WAVES   4          // waves per block
#define ROWPAD  260        // 256 + 4 dwords: bank offset 4*row -> conflict-free ds_load_b64 reads
#define EPS     1e-12f

// ---------------------------------------------------------------------------
// Kernel 1: normalize the [G,256] query matrix into workspace.
// One block per graph, 256 threads, block tree-reduction for the norm.
// ---------------------------------------------------------------------------
__global__ __launch_bounds__(256) void qnorm_kernel(const float* __restrict__ q,
                                                    float* __restrict__ qn) {
  __shared__ float red[256];
  const int g = blockIdx.x;
  const int t = threadIdx.x;
  const float v = q[g * HIDDEN + t];
  red[t] = v * v;
  __syncthreads();
  for (int s = 128; s > 0; s >>= 1) {
    if (t < s) red[t] += red[t + s];
    __syncthreads();
  }
  const float n = fmaxf(sqrtf(red[0]), EPS);
  qn[g * HIDDEN + t] = v / n;
}

// ---------------------------------------------------------------------------
// Kernel 2: per-node cosine score via V_WMMA_F32_16X16X4_F32.
// Each wave owns a 16-node tile:
//   A      = x-tile (16 x K)               (f32 A-fragment from LDS)
//   B_q    = gathered normalized queries   (same fragment layout, from LDS)
//   wmma(a, a)   -> Gram matrix, diag = ||x_n||^2
//   wmma(a, b_q) -> diag = dot(x_n, qhat_{b[n]})
// Staging uses CDNA5 async global->LDS copies (ASYNCcnt path): the memory
// pipe writes LDS directly, no VGPR round trip, one s_wait_asynccnt 0.
// ---------------------------------------------------------------------------
__global__ __launch_bounds__(WAVES * 32) void score_kernel(
    const float* __restrict__ x, const float* __restrict__ qn,
    const int* __restrict__ bidx, float* __restrict__ out, int N) {
  __shared__ __align__(16) float xs[WAVES][16][ROWPAD];
  __shared__ __align__(16) float qs[WAVES][16][ROWPAD];
  __shared__ int gsh[WAVES][16];

  const int lane = threadIdx.x & 31;
  const int wv   = threadIdx.x >> 5;
  const long tileBase = (long)blockIdx.x * (WAVES * 16) + (long)wv * 16;

  // Stage the 16 graph ids for this wave's tile (clamped for the tail tile;
  // WMMA requires EXEC all-ones, so we never mask compute, only the store).
  if (lane < 16) {
    long node = tileBase + lane;
    if (node >= N) node = N - 1;
    gsh[wv][lane] = bidx[node];
  }
  __syncthreads();

  // Async-stage x tile and gathered q tile: 32 iters, each lane issues two
  // global_load_async_to_lds_b128 (16B). Coalesced: a wave covers half a row
  // (32 lanes * 4 floats = 128 contiguous floats) per source per iteration.
  for (int it = 0; it < 32; ++it) {
    const int e4  = it * 32 + lane;      // float4 index within the tile
    const int row = e4 >> 6;             // 64 float4 per row
    const int c4  = e4 & 63;
    long node = tileBase + row;
    if (node >= N) node = N - 1;
    const float* gx = x + node * HIDDEN + (long)c4 * 4;
    const int g = gsh[wv][row];
    const float* gq = qn + (long)g * HIDDEN + (long)c4 * 4;
    // LDS aperture maps flat addr[31:0] straight to the LDS byte offset.
    unsigned lx = (unsigned)(uintptr_t)&xs[wv][row][c4 * 4];
    unsigned lq = (unsigned)(uintptr_t)&qs[wv][row][c4 * 4];
    asm volatile(
        "global_load_async_to_lds_b128 %0, %2, off\n\t"
        "global_load_async_to_lds_b128 %1, %3, off"
        :
        : "v"(lx), "v"(lq), "v"(gx), "v"(gq)
        : "memory");
  }
  asm volatile("s_wait_asynccnt 0x0" ::: "memory");
  __syncthreads();

  // f32 WMMA fragment addressing: lane L reads (row = L%16, k + 2*(L/16)).
  const int row   = lane & 15;
  const int khalf = (lane >> 4) << 1;    // 0 or 2
  const float* xrow = &xs[wv][row][khalf];
  const float* qrow = &qs[wv][row][khalf];

  v8f cdot = {0.f, 0.f, 0.f, 0.f, 0.f, 0.f, 0.f, 0.f};
  v8f cnrm = {0.f, 0.f, 0.f, 0.f, 0.f, 0.f, 0.f, 0.f};

#pragma unroll 8
  for (int k = 0; k < HIDDEN; k += 4) {
    v2f a = *(const v2f*)(xrow + k);     // ds_load_b64, bank-conflict-free
    v2f b = *(const v2f*)(qrow + k);
    // D = A*A^T accumulate: diagonal -> squared norms
    cnrm = __builtin_amdgcn_wmma_f32_16x16x4_f32(
        false, a, false, a, (short)0, cnrm, false, false);
    // D = A*Bq accumulate: diagonal -> dot(x_n, qhat_n)
    cdot = __builtin_amdgcn_wmma_f32_16x16x4_f32(
        false, a, false, b, (short)0, cdot, false, false);
  }

  // Diagonal extraction from the 16x16 f32 C/D layout:
  //   n in 0..7  -> VGPR n,   lane n        (lanes 0-7,  c[lane])
  //   n in 8..15 -> VGPR n-8, lane n+16     (lanes 24-31, c[lane-24])
  const int i = lane & 7;
  float dv = cdot[0];
  float nv = cnrm[0];
#pragma unroll
  for (int j = 1; j < 8; ++j) {
    if (i == j) { dv = cdot[j]; nv = cnrm[j]; }
  }
  const bool sel = (lane < 8) || (lane >= 24);
  const long n = tileBase + ((lane < 8) ? lane : lane - 16);
  if (sel && n < N) {
    const float len = fmaxf(sqrtf(nv), EPS);
    out[n] = dv / len;
  }
}

// ---------------------------------------------------------------------------
extern "C" void kernel_launch(void* const* d_in, const int* in_sizes, int n_in,
                              void* d_out, int out_size, void* d_ws, size_t ws_size,
                              hipStream_t stream) {
  const float* x    = (const float*)d_in[0];   // [N, 256] f32
  const float* q    = (const float*)d_in[1];   // [G, 256] f32
  const int*   bidx = (const int*)d_in[2];     // [N] int32 (sorted)
  float* out = (float*)d_out;                  // [N] f32

  const int N = in_sizes[0] / HIDDEN;
  const int G = in_sizes[1] / HIDDEN;

  float* qn = (float*)d_ws;                    // [G, 256] normalized queries

  qnorm_kernel<<<G, 256, 0, stream>>>(q, qn);

  const int nodesPerBlock = WAVES * 16;
  const int blocks = (N + nodesPerBlock - 1) / nodesPerBlock;
  score_kernel<<<blocks, WAVES * 32, 0, stream>>>(x, qn, bidx, out, N);
}